// aspect_rating_1_46188078301357
// MI455X (gfx1250) — compile-verified
//
#include <hip/hip_runtime.h>
#include <math.h>

// ---------------- problem constants (from reference) ----------------
constexpr int VOCAB = 32000, DD = 200, AA = 20, NEG = 2;
constexpr int RR = 4096, LL = 128, BB = 1024, NNZ = 20480;

// ---------------- output layout (flat concat, floats) ----------------
constexpr int OBJ_OFF  = 0;
constexpr int RL_OFF   = 1;
constexpr int ABAE_OFF = RL_OFF + BB;            // 1025
constexpr int PRED_OFF = ABAE_OFF + RR * NEG;    // 9217
constexpr int UEMB_OFF = PRED_OFF + BB;          // 10241
constexpr int IEMB_OFF = UEMB_OFF + BB * DD;     // 215041

// ---------------- workspace layout (floats) ----------------
constexpr int WS_ZS    = 0;                      // R*D
constexpr int WS_PT    = WS_ZS + RR * DD;        // R*20
constexpr int WS_RS    = WS_PT + RR * AA;        // R*D
constexpr int WS_QUAD  = WS_RS + RR * DD;        // B
constexpr int WS_LIN   = WS_QUAD + BB;           // B
constexpr int WS_JSUM  = WS_LIN + BB;            // 1
constexpr int WS_ULOSS = WS_JSUM + 1;            // 1
constexpr int WS_TQUAD = WS_ULOSS + 1;           // 1
constexpr int WS_WWP   = WS_TQUAD + 3;           // 32*200  (odd base)
constexpr int WS_WTP   = WS_WWP + 32 * DD;       // 208*20
constexpr int WS_FMVT  = WS_WTP + 208 * AA;      // 16*400

typedef float v2f __attribute__((ext_vector_type(2)));
typedef float v8f __attribute__((ext_vector_type(8)));

__device__ inline v8f wmma_f32(v2f a, v2f b, v8f c) {
  // V_WMMA_F32_16X16X4_F32 : D = A(16x4) x B(4x16) + C(16x16), all f32
  return __builtin_amdgcn_wmma_f32_16x16x4_f32(false, a, false, b, (short)0, c,
                                               false, false);
}
__device__ inline v2f ld2(const float* p) { return *(const v2f*)p; }

// =====================================================================
// Kernel 0: pad the small B-operands once so WMMA inner loops have
// unconditional b64 loads (no EXEC-masked load blocks).
//   WwP  : (32 x 200)  rows n<20  = Ww[n][k]   else 0   ([n][k])
//   WtP  : (208 x 20)  rows n<200 = Wt[n][k]   else 0   ([n][k])
//   fmVT : (16 x 400)  rows n<10  = fm_V[k][n] else 0   ([n][k])
// =====================================================================
__global__ __launch_bounds__(256) void k_prep(
    const float* __restrict__ Ww, const float* __restrict__ Wt,
    const float* __restrict__ fmV, float* __restrict__ WwP,
    float* __restrict__ WtP, float* __restrict__ fmVT) {
  const int i = blockIdx.x * 256 + threadIdx.x;
  if (i < 32 * DD) {
    const int n = i / DD, k = i - n * DD;
    WwP[i] = (n < AA) ? Ww[n * DD + k] : 0.f;
  }
  const int j = i - 32 * DD;
  if (j >= 0 && j < 208 * AA) {
    const int n = j / AA, k = j - n * AA;
    WtP[j] = (n < DD) ? Wt[n * AA + k] : 0.f;
  }
  const int h = j - 208 * AA;
  if (h >= 0 && h < 16 * 2 * DD) {
    const int n = h / (2 * DD), k = h - n * (2 * DD);
    fmVT[h] = (n < 10) ? fmV[k * 10 + n] : 0.f;
  }
}

// =====================================================================
// Kernel 1: per-review ABAE attention. One block per review.
// Stages e_w (128x200 f32 = 102.4KB) in LDS once, reused 3x
// (y_s mean, dx matvec, raw-reshape z_s). Gather-bound; word_emb is
// L2-resident (25.6MB << 192MB L2).
// =====================================================================
__global__ __launch_bounds__(256) void k_attention(
    const int* __restrict__ hist, const float* __restrict__ word_emb,
    const float* __restrict__ Wm, float* __restrict__ zs) {
  extern __shared__ unsigned char smem_raw[];
  float* e   = (float*)smem_raw;        // 25600 floats, flat (L,D) row-major
  float* ys  = e + LL * DD;             // 200
  float* v   = ys + DD;                 // 200
  float* ax  = v + DD;                  // 128
  float* red = ax + LL;                 // 256
  int*   tok = (int*)(red + 256);       // 128

  const int r = blockIdx.x;
  const int t = threadIdx.x;

  if (t < LL) tok[t] = hist[r * LL + t];
  __syncthreads();

  // stage e_w tile
  for (int j = t; j < LL * DD; j += 256) {
    const int l = j / DD, d = j - l * DD;
    e[j] = word_emb[tok[l] * DD + d];
  }
  __syncthreads();

  // y_s = mean over L
  if (t < DD) {
    float s = 0.f;
    for (int l = 0; l < LL; ++l) s += e[l * DD + t];
    ys[t] = s * (1.0f / LL);
  }
  __syncthreads();

  // v = Wm^T y_s  (factored form of einsum('rld,rd->rl', e_w@Wm.T, y_s))
  if (t < DD) {
    float s = 0.f;
    for (int d = 0; d < DD; ++d) s += Wm[d * DD + t] * ys[d];
    v[t] = s;
  }
  __syncthreads();

  // dx[l] = e_w[l] . v ; softmax over l
  float dxl = 0.f;
  if (t < LL) {
    float s = 0.f;
    for (int d = 0; d < DD; ++d) s += e[t * DD + d] * v[d];
    dxl = s;
  }
  red[t] = (t < LL) ? dxl : -1e30f;
  __syncthreads();
  for (int off = 128; off >= 1; off >>= 1) {
    if (t < off) red[t] = fmaxf(red[t], red[t + off]);
    __syncthreads();
  }
  const float mx = red[0];
  __syncthreads();
  const float ex = (t < LL) ? expf(dxl - mx) : 0.f;
  red[t] = ex;
  __syncthreads();
  for (int off = 128; off >= 1; off >>= 1) {
    if (t < off) red[t] += red[t + off];
    __syncthreads();
  }
  const float inv = 1.0f / red[0];
  if (t < LL) ax[t] = ex * inv;
  __syncthreads();

  // z_s via the *raw reshape* (R,L,D)->(R,D,L): z_s[d] = sum_l flat[d*L+l]*ax[l]
  if (t < DD) {
    float s = 0.f;
    const float* ef = e + t * LL;
    for (int l = 0; l < LL; ++l) s += ef[l] * ax[l];
    zs[r * DD + t] = s;
  }
}

// =====================================================================
// Kernel 2 (WMMA): p_t = z_s @ Ww^T + bw  -> (4096, 20)
// One wave per 16x16 output tile; K=200 in 50 steps of 4.
// Inner loop: 2x global_load_b64 + v_wmma, fully unconditional.
// =====================================================================
__global__ __launch_bounds__(32) void k_pt_wmma(
    const float* __restrict__ zs, const float* __restrict__ WwP,
    const float* __restrict__ bw, float* __restrict__ pt) {
  const int mt = blockIdx.x, nt = blockIdx.y;
  const int lane = threadIdx.x;
  const int half = lane >> 4, lid = lane & 15;
  const float* pa = zs + (mt * 16 + lid) * DD;
  const float* pb = WwP + (nt * 16 + lid) * DD;
  v8f acc = {};
  for (int k0 = 0; k0 < DD; k0 += 4) {
    const int ka = k0 + half * 2;
    acc = wmma_f32(ld2(pa + ka), ld2(pb + ka), acc);
  }
  const int col = nt * 16 + lid;
  if (col < AA) {
    const float bias = bw[col];
#pragma unroll
    for (int i = 0; i < 8; ++i) {
      const int row = mt * 16 + i + half * 8;
      pt[row * AA + col] = acc[i] + bias;
    }
  }
}

// =====================================================================
// Kernel 3 (WMMA): r_s = p_t @ Wt^T -> (4096, 200). K=20 (5 steps).
// =====================================================================
__global__ __launch_bounds__(32) void k_rs_wmma(
    const float* __restrict__ pt, const float* __restrict__ WtP,
    float* __restrict__ rs) {
  const int mt = blockIdx.x, nt = blockIdx.y;  // nt 0..12 (208 cols padded)
  const int lane = threadIdx.x;
  const int half = lane >> 4, lid = lane & 15;
  const float* pa = pt + (mt * 16 + lid) * AA;
  const float* pb = WtP + (nt * 16 + lid) * AA;
  v8f acc = {};
#pragma unroll
  for (int k0 = 0; k0 < AA; k0 += 4) {
    const int ka = k0 + half * 2;
    acc = wmma_f32(ld2(pa + ka), ld2(pb + ka), acc);
  }
  const int col = nt * 16 + lid;
  if (col < DD) {
#pragma unroll
    for (int i = 0; i < 8; ++i)
      rs[(mt * 16 + i + half * 8) * DD + col] = acc[i];
  }
}

// =====================================================================
// Kernel 4: margin ranking loss per (r, neg). One block per rn.
// =====================================================================
__global__ __launch_bounds__(256) void k_abae(
    const int* __restrict__ neg, const float* __restrict__ word_emb,
    const float* __restrict__ zs, const float* __restrict__ rs,
    float* __restrict__ out_abae) {
  __shared__ int tok[LL];
  __shared__ float red[5][256];
  const int rn = blockIdx.x;
  const int r = rn / NEG;
  const int t = threadIdx.x;
  if (t < LL) tok[t] = neg[rn * LL + t];
  __syncthreads();

  float zn = 0.f, rsv = 0.f, zsv = 0.f;
  if (t < DD) {
    float s = 0.f;
    for (int l = 0; l < LL; ++l) s += word_emb[tok[l] * DD + t];
    zn = s * (1.0f / LL);
    rsv = rs[r * DD + t];
    zsv = zs[r * DD + t];
  }
  red[0][t] = zn * zn;
  red[1][t] = rsv * rsv;
  red[2][t] = zsv * zsv;
  red[3][t] = zn * rsv;
  red[4][t] = rsv * zsv;
  __syncthreads();
  for (int off = 128; off >= 1; off >>= 1) {
    if (t < off)
      for (int q = 0; q < 5; ++q) red[q][t] += red[q][t + off];
    __syncthreads();
  }
  if (t == 0) {
    const float nzn = fmaxf(sqrtf(red[0][0]), 1e-12f);
    const float nr  = fmaxf(sqrtf(red[1][0]), 1e-12f);
    const float nzs = fmaxf(sqrtf(red[2][0]), 1e-12f);
    const float c2 = red[3][0] / (nzn * nr);
    const float c1 = red[4][0] / (nr * nzs);
    out_abae[rn] = fmaxf(c2 - c1 + 1.0f, 0.f);
  }
}

// ---------------- small reductions ----------------
__global__ __launch_bounds__(256) void k_reduce_j(const float* __restrict__ x,
                                                  float* __restrict__ out) {
  __shared__ float red[256];
  const int t = threadIdx.x;
  float s = 0.f;
  for (int i = t; i < RR * NEG; i += 256) s += x[i];
  red[t] = s;
  __syncthreads();
  for (int off = 128; off >= 1; off >>= 1) {
    if (t < off) red[t] += red[t + off];
    __syncthreads();
  }
  if (t == 0) out[0] = red[0];
}

__global__ __launch_bounds__(256) void k_uloss(const float* __restrict__ Wt,
                                               float* __restrict__ uloss) {
  __shared__ float nrm[AA];
  __shared__ float red[256];
  const int t = threadIdx.x;
  if (t < AA) {
    float s = 0.f;
    for (int d = 0; d < DD; ++d) { const float w = Wt[d * AA + t]; s += w * w; }
    nrm[t] = fmaxf(sqrtf(s), 1e-12f);
  }
  __syncthreads();
  float acc = 0.f;
  for (int p = t; p < AA * AA; p += 256) {
    const int i = p / AA, j = p - i * AA;
    float g = 0.f;
    for (int d = 0; d < DD; ++d) g += Wt[d * AA + i] * Wt[d * AA + j];
    g /= (nrm[i] * nrm[j]);
    const float diff = g - (i == j ? 1.f : 0.f);
    acc += diff * diff;
  }
  red[t] = acc;
  __syncthreads();
  for (int off = 128; off >= 1; off >>= 1) {
    if (t < off) red[t] += red[t + off];
    __syncthreads();
  }
  if (t == 0) uloss[0] = red[0] / (float)(AA * AA);
}

// =====================================================================
// Kernel 7: deterministic sparse (B,R)@(R,D) aggregation.
// One block per output row (user rows then item rows); indices staged
// through LDS in chunks; no float atomics -> bitwise reproducible.
// =====================================================================
constexpr int CHUNK = 2048;
__global__ __launch_bounds__(256) void k_aspect(
    const int* __restrict__ u_idx, const float* __restrict__ u_val,
    const int* __restrict__ i_idx, const float* __restrict__ i_val,
    const float* __restrict__ rs, float* __restrict__ uemb,
    float* __restrict__ iemb) {
  __shared__ int rows_c[CHUNK];
  __shared__ int cols_c[CHUNK];
  __shared__ float vals_c[CHUNK];
  const int blk = blockIdx.x;
  const bool is_item = blk >= BB;
  const int b = is_item ? blk - BB : blk;
  const int* rows = is_item ? i_idx : u_idx;
  const int* cols = rows + NNZ;
  const float* vals = is_item ? i_val : u_val;
  float* out = is_item ? iemb : uemb;
  const int t = threadIdx.x;
  float acc = 0.f;
  for (int base = 0; base < NNZ; base += CHUNK) {
    for (int j = t; j < CHUNK; j += 256) {
      rows_c[j] = rows[base + j];
      cols_c[j] = cols[base + j];
      vals_c[j] = vals[base + j];
    }
    __syncthreads();
    if (t < DD) {
      for (int j = 0; j < CHUNK; ++j)
        if (rows_c[j] == b) acc += vals_c[j] * rs[cols_c[j] * DD + t];
    }
    __syncthreads();
  }
  if (t < DD) out[b * DD + t] = acc;
}

// =====================================================================
// Kernel 8 (WMMA): FM interaction. vec@V and vec^2@V^2 as two
// accumulators over K=400 (100 steps); per-row quad + linear term.
// A-operand pointer is select-ed (v_cndmask on address), loads stay
// unconditional; B comes from the padded transposed fm_V.
// =====================================================================
__global__ __launch_bounds__(32) void k_fm_wmma(
    const float* __restrict__ uemb, const float* __restrict__ iemb,
    const float* __restrict__ fmVT, const float* __restrict__ fcw,
    const float* __restrict__ fcb, float* __restrict__ quad,
    float* __restrict__ lin) {
  __shared__ float s1[16][17];
  __shared__ float s2[16][17];
  const int mt = blockIdx.x;  // 0..63
  const int lane = threadIdx.x;
  const int half = lane >> 4, lid = lane & 15;
  const int m = mt * 16 + lid;
  const float* um = uemb + m * DD;
  const float* im = iemb + m * DD;
  const float* pb = fmVT + lid * (2 * DD);
  v8f a1 = {}, a2 = {};
  for (int k0 = 0; k0 < 2 * DD; k0 += 4) {
    const int ka = k0 + half * 2;
    const float* src = (ka < DD) ? (um + ka) : (im + (ka - DD));
    const float x0 = src[0];
    const float x1 = src[1];
    const v2f b = ld2(pb + ka);
    const v2f a = {x0, x1};
    a1 = wmma_f32(a, b, a1);
    const v2f aq = {x0 * x0, x1 * x1};
    const v2f bq = {b.x * b.x, b.y * b.y};
    a2 = wmma_f32(aq, bq, a2);
  }
#pragma unroll
  for (int i = 0; i < 8; ++i) {
    s1[i + half * 8][lid] = a1[i];
    s2[i + half * 8][lid] = a2[i];
  }
  __syncthreads();
  if (lane < 16) {
    const int row = lane;
    float q = 0.f;
    for (int n = 0; n < 10; ++n) {
      const float s = s1[row][n];
      q += s * s - s2[row][n];
    }
    const int bidx = mt * 16 + row;
    quad[bidx] = q;
    float lv = fcb[0];
    for (int j = 0; j < DD; ++j) lv += uemb[bidx * DD + j] * fcw[j];
    for (int j = 0; j < DD; ++j) lv += iemb[bidx * DD + j] * fcw[DD + j];
    lin[bidx] = lv;
  }
}

__global__ __launch_bounds__(256) void k_reduce_q(const float* __restrict__ q,
                                                  float* __restrict__ out) {
  __shared__ float red[256];
  const int t = threadIdx.x;
  float s = 0.f;
  for (int i = t; i < BB; i += 256) s += q[i];
  red[t] = s;
  __syncthreads();
  for (int off = 128; off >= 1; off >>= 1) {
    if (t < off) red[t] += red[t + off];
    __syncthreads();
  }
  if (t == 0) out[0] = red[0];
}

__global__ __launch_bounds__(256) void k_pred(
    const float* __restrict__ quadtot, const float* __restrict__ lin,
    const int* __restrict__ user, const int* __restrict__ item,
    const float* __restrict__ busers, const float* __restrict__ bitems,
    const float* __restrict__ label, float* __restrict__ pred,
    float* __restrict__ rl) {
  const int i = blockIdx.x * 256 + threadIdx.x;
  if (i < BB) {
    const float p =
        0.5f * quadtot[0] + lin[i] + busers[user[i]] + bitems[item[i]];
    pred[i] = p;
    const float d = p - label[i];
    rl[i] = d * d;
  }
}

__global__ __launch_bounds__(256) void k_final(const float* __restrict__ rl,
                                               const float* __restrict__ jsum,
                                               const float* __restrict__ uloss,
                                               float* __restrict__ obj) {
  __shared__ float red[256];
  const int t = threadIdx.x;
  float s = 0.f;
  for (int i = t; i < BB; i += 256) s += rl[i];
  red[t] = s;
  __syncthreads();
  for (int off = 128; off >= 1; off >>= 1) {
    if (t < off) red[t] += red[t + off];
    __syncthreads();
  }
  if (t == 0)
    obj[0] = red[0] / (float)BB + 0.01f * (jsum[0] / (float)(RR * NEG)) +
             0.01f * uloss[0];
}

// =====================================================================
extern "C" void kernel_launch(void* const* d_in, const int* in_sizes, int n_in,
                              void* d_out, int out_size, void* d_ws,
                              size_t ws_size, hipStream_t stream) {
  const int*   hist     = (const int*)d_in[0];
  const int*   neg      = (const int*)d_in[1];
  const int*   user     = (const int*)d_in[2];
  const int*   item     = (const int*)d_in[3];
  const float* label    = (const float*)d_in[4];
  const int*   u_idx    = (const int*)d_in[5];
  const float* u_val    = (const float*)d_in[6];
  const int*   i_idx    = (const int*)d_in[7];
  const float* i_val    = (const float*)d_in[8];
  const float* word_emb = (const float*)d_in[9];
  const float* Wm       = (const float*)d_in[10];
  const float* Ww       = (const float*)d_in[11];
  const float* bw       = (const float*)d_in[12];
  const float* Wt       = (const float*)d_in[13];
  const float* fcw      = (const float*)d_in[14];
  const float* fcb      = (const float*)d_in[15];
  const float* fmV      = (const float*)d_in[16];
  const float* busers   = (const float*)d_in[17];
  const float* bitems   = (const float*)d_in[18];

  float* out = (float*)d_out;
  float* ws  = (float*)d_ws;

  const size_t smem1 =
      (size_t)(LL * DD + DD + DD + LL + 256) * sizeof(float) +
      (size_t)LL * sizeof(int);  // ~106 KB: allow >64KB dynamic LDS
  (void)hipFuncSetAttribute((const void*)k_attention,
                            hipFuncAttributeMaxDynamicSharedMemorySize,
                            (int)smem1);

  // 0. pad B-operands for the WMMA GEMMs
  k_prep<<<67, 256, 0, stream>>>(Ww, Wt, fmV, ws + WS_WWP, ws + WS_WTP,
                                 ws + WS_FMVT);
  // 1. attention -> z_s
  k_attention<<<RR, 256, smem1, stream>>>(hist, word_emb, Wm, ws + WS_ZS);
  // 2. p_t = z_s @ Ww^T + bw    (WMMA f32 16x16x4)
  k_pt_wmma<<<dim3(RR / 16, 2), 32, 0, stream>>>(ws + WS_ZS, ws + WS_WWP, bw,
                                                 ws + WS_PT);
  // 3. r_s = p_t @ Wt^T         (WMMA f32 16x16x4)
  k_rs_wmma<<<dim3(RR / 16, 13), 32, 0, stream>>>(ws + WS_PT, ws + WS_WTP,
                                                  ws + WS_RS);
  // 4. margin ranking losses
  k_abae<<<RR * NEG, 256, 0, stream>>>(neg, word_emb, ws + WS_ZS, ws + WS_RS,
                                       out + ABAE_OFF);
  // 5. J loss sum
  k_reduce_j<<<1, 256, 0, stream>>>(out + ABAE_OFF, ws + WS_JSUM);
  // 6. orthogonality regularizer
  k_uloss<<<1, 256, 0, stream>>>(Wt, ws + WS_ULOSS);
  // 7. sparse aggregation -> user/item aspect embeds (direct to d_out)
  k_aspect<<<2 * BB, 256, 0, stream>>>(u_idx, u_val, i_idx, i_val, ws + WS_RS,
                                       out + UEMB_OFF, out + IEMB_OFF);
  // 8. FM quadratic + linear    (WMMA f32 16x16x4, dual accumulators)
  k_fm_wmma<<<BB / 16, 32, 0, stream>>>(out + UEMB_OFF, out + IEMB_OFF,
                                        ws + WS_FMVT, fcw, fcb, ws + WS_QUAD,
                                        ws + WS_LIN);
  // 9. global quadratic sum (faithful-to-source scalar)
  k_reduce_q<<<1, 256, 0, stream>>>(ws + WS_QUAD, ws + WS_TQUAD);
  // 10. prediction + rating loss
  k_pred<<<BB / 256, 256, 0, stream>>>(ws + WS_TQUAD, ws + WS_LIN, user, item,
                                       busers, bitems, label, out + PRED_OFF,
                                       out + RL_OFF);
  // 11. obj loss
  k_final<<<1, 256, 0, stream>>>(out + RL_OFF, ws + WS_JSUM, ws + WS_ULOSS,
                                 out + OBJ_OFF);
}